// MixtureTokenModule_8151847928238
// MI455X (gfx1250) — compile-verified
//
#include <hip/hip_runtime.h>
#include <math.h>

#define Bn    1024
#define Sn    4096
#define Fn    5
#define Kn    3
#define DESC1 34        // 33 descriptor values + valid flag (= GEMM1 K dim)
#define PH    128
#define HIDn  4096
#define TPB   256
#define NWAVE (TPB / 32)
#define EMITERS 100
#define REGC  1e-4f

typedef __attribute__((ext_vector_type(2))) float v2f;
typedef __attribute__((ext_vector_type(8))) float v8f;

// Deterministic block-wide sum of vals[0..nv-1]: wave32 shuffle tree, then a
// fixed-order sum of the 8 per-wave partials (same result on every replay).
__device__ __forceinline__ void block_reduce_arr(float* vals, int nv, float* wred,
                                                 int lane, int wave) {
  for (int i = 0; i < nv; ++i) {
    float v = vals[i];
    v += __shfl_down(v, 16, 32);
    v += __shfl_down(v, 8, 32);
    v += __shfl_down(v, 4, 32);
    v += __shfl_down(v, 2, 32);
    v += __shfl_down(v, 1, 32);
    if (lane == 0) wred[wave * nv + i] = v;
  }
  __syncthreads();
  for (int i = 0; i < nv; ++i) {
    float t = 0.f;
    for (int w = 0; w < NWAVE; ++w) t += wred[w * nv + i];
    vals[i] = t;
  }
  __syncthreads();
}

// ---------------------------------------------------------------------------
// Kernel 1: one workgroup per sample. feats staged in LDS once; 100 EM iters.
// Dynamic LDS: xs[S*F] + mk[S] + rr[S*K] + wred[8*18]  (~145 KB, CDNA5-only)
// ---------------------------------------------------------------------------
__global__ __launch_bounds__(TPB)
void gmm_em_kernel(const float* __restrict__ fix, const int* __restrict__ attn,
                   float* __restrict__ desc) {
  extern __shared__ float lds[];
  float* xs   = lds;                 // Sn*Fn
  float* mk   = xs + Sn * Fn;        // Sn
  float* rr   = mk + Sn;             // Sn*Kn
  float* wred = rr + Sn * Kn;        // NWAVE*18
  const int b = blockIdx.x;
  const int tid = threadIdx.x;
  const int lane = tid & 31, wave = tid >> 5;

  // load + log1p transform + mask
  for (int s = tid; s < Sn; s += TPB) {
    float nrm2 = 0.f;
    #pragma unroll
    for (int f = 0; f < Fn; ++f) {
      float t = fix[((size_t)b * Sn + s) * Fn + f];
      t = log1pf(fmaxf(t, 0.f));
      xs[s * Fn + f] = t;
      nrm2 += t * t;
    }
    mk[s] = ((attn[(size_t)b * Sn + s] != 0) && (sqrtf(nrm2) > 1e-6f)) ? 1.f : 0.f;
  }
  __syncthreads();

  // global moments for init
  float st[1 + Fn];
  #pragma unroll
  for (int i = 0; i <= Fn; ++i) st[i] = 0.f;
  for (int s = tid; s < Sn; s += TPB) {
    float m = mk[s];
    st[0] += m;
    #pragma unroll
    for (int f = 0; f < Fn; ++f) st[1 + f] += m * xs[s * Fn + f];
  }
  block_reduce_arr(st, 1 + Fn, wred, lane, wave);
  const float msum = st[0];
  const float n = fmaxf(msum, 1.f);
  float gmean[Fn];
  #pragma unroll
  for (int f = 0; f < Fn; ++f) gmean[f] = st[1 + f] / n;

  float sv[Fn];
  #pragma unroll
  for (int f = 0; f < Fn; ++f) sv[f] = 0.f;
  for (int s = tid; s < Sn; s += TPB) {
    float m = mk[s];
    #pragma unroll
    for (int f = 0; f < Fn; ++f) {
      float d = xs[s * Fn + f] - gmean[f];
      sv[f] += m * d * d;
    }
  }
  block_reduce_arr(sv, Fn, wred, lane, wave);

  float w[Kn], mu[Kn][Fn], var[Kn][Fn];
  #pragma unroll
  for (int f = 0; f < Fn; ++f) sv[f] = sv[f] / n + REGC;
  #pragma unroll
  for (int k = 0; k < Kn; ++k) {
    w[k] = 1.f / 3.f;
    #pragma unroll
    for (int f = 0; f < Fn; ++f) {
      mu[k][f]  = gmean[f] + (float)(k - 1) * sqrtf(sv[f]);  // linspace(-1,0,1)
      var[k][f] = sv[f];
    }
  }

  const float log2pi = 1.8378770664093453f;
  for (int it = 0; it < EMITERS; ++it) {
    float base[Kn];
    #pragma unroll
    for (int k = 0; k < Kn; ++k) {
      float slv = 0.f;
      #pragma unroll
      for (int f = 0; f < Fn; ++f) slv += logf(var[k][f]);
      base[k] = logf(w[k]) - 0.5f * (slv + (float)Fn * log2pi);
    }
    // E-step + weighted sums
    float acc[Kn + Kn * Fn];
    #pragma unroll
    for (int i = 0; i < Kn + Kn * Fn; ++i) acc[i] = 0.f;
    for (int s = tid; s < Sn; s += TPB) {
      float xv[Fn];
      #pragma unroll
      for (int f = 0; f < Fn; ++f) xv[f] = xs[s * Fn + f];
      float lp[Kn];
      #pragma unroll
      for (int k = 0; k < Kn; ++k) {
        float q = 0.f;
        #pragma unroll
        for (int f = 0; f < Fn; ++f) {
          float d = xv[f] - mu[k][f];
          q += d * d / var[k][f];
        }
        lp[k] = base[k] - 0.5f * q;
      }
      float mx = fmaxf(lp[0], fmaxf(lp[1], lp[2]));
      float e[Kn], se = 0.f;
      #pragma unroll
      for (int k = 0; k < Kn; ++k) { e[k] = expf(lp[k] - mx); se += e[k]; }
      float sc = mk[s] / se;
      #pragma unroll
      for (int k = 0; k < Kn; ++k) {
        float r = e[k] * sc;
        rr[s * Kn + k] = r;
        acc[k] += r;
        #pragma unroll
        for (int f = 0; f < Fn; ++f) acc[Kn + k * Fn + f] += r * xv[f];
      }
    }
    block_reduce_arr(acc, Kn + Kn * Fn, wred, lane, wave);

    float Nk[Kn], w2[Kn], mu2[Kn][Fn];
    #pragma unroll
    for (int k = 0; k < Kn; ++k) {
      Nk[k] = acc[k] + 1e-10f;
      w2[k] = Nk[k] / n;
      #pragma unroll
      for (int f = 0; f < Fn; ++f) mu2[k][f] = acc[Kn + k * Fn + f] / Nk[k];
    }
    // variance pass (uses stored responsibilities)
    float a2[Kn * Fn];
    #pragma unroll
    for (int i = 0; i < Kn * Fn; ++i) a2[i] = 0.f;
    for (int s = tid; s < Sn; s += TPB) {
      float r0 = rr[s * Kn + 0], r1 = rr[s * Kn + 1], r2 = rr[s * Kn + 2];
      #pragma unroll
      for (int f = 0; f < Fn; ++f) {
        float x = xs[s * Fn + f];
        float d0 = x - mu2[0][f], d1 = x - mu2[1][f], d2 = x - mu2[2][f];
        a2[0 * Fn + f] += r0 * d0 * d0;
        a2[1 * Fn + f] += r1 * d1 * d1;
        a2[2 * Fn + f] += r2 * d2 * d2;
      }
    }
    block_reduce_arr(a2, Kn * Fn, wred, lane, wave);
    #pragma unroll
    for (int k = 0; k < Kn; ++k) {
      w[k] = w2[k];
      #pragma unroll
      for (int f = 0; f < Fn; ++f) {
        mu[k][f]  = mu2[k][f];
        var[k][f] = a2[k * Fn + f] / Nk[k] + REGC;
      }
    }
  }

  if (tid == 0) {
    int o[3] = {0, 1, 2};               // stable argsort(-w): strict-> swap only
    for (int i = 0; i < 2; ++i)
      for (int j = 0; j < 2 - i; ++j)
        if (w[o[j + 1]] > w[o[j]]) { int t = o[j]; o[j] = o[j + 1]; o[j + 1] = t; }
    float d[DESC1];
    for (int k = 0; k < Kn; ++k) {
      d[k] = w[o[k]];
      for (int f = 0; f < Fn; ++f) {
        d[3 + k * Fn + f]  = mu[o[k]][f];
        d[18 + k * Fn + f] = var[o[k]][f];
      }
    }
    bool ok = (msum >= 6.0f);           // MIN_PTS = max(2K,4) = 6
    float mabs = 0.f;
    for (int i = 0; i < 33; ++i) {
      if (!ok) d[i] = 0.f;
      mabs = fmaxf(mabs, fabsf(d[i]));
    }
    d[33] = (mabs > 1e-8f) ? 1.f : 0.f;
    for (int i = 0; i < DESC1; ++i) desc[b * DESC1 + i] = d[i];
  }
}

// ---------------------------------------------------------------------------
// Kernel 2: GEMM1 (1024x34 @ 34x128) via f32 WMMA 16x16x4 + LN + erf-GELU.
// 64 blocks x 8 waves; wave w owns N-tile w; K padded 34->36 (9 wmma steps).
// ---------------------------------------------------------------------------
__global__ __launch_bounds__(TPB)
void mlp1_kernel(const float* __restrict__ desc, const float* __restrict__ W1,
                 const float* __restrict__ b1, const float* __restrict__ g1,
                 const float* __restrict__ bb1, float* __restrict__ h) {
  __shared__ float hout[16 * PH];
  __shared__ float mrow[16], rstd[16];
  const int tid = threadIdx.x;
  const int lane = tid & 31, wave = tid >> 5;
  const int nl = lane & 15, kh = lane >> 4;
  const int rb = blockIdx.x * 16;
  const int nn = wave * 16 + nl;

  v8f c = {0.f, 0.f, 0.f, 0.f, 0.f, 0.f, 0.f, 0.f};
  for (int kb = 0; kb < 36; kb += 4) {
    int ka = kb + 2 * kh;
    v2f a, bv;
    // A frag: lane holds A[m=nl][ka], A[m=nl][ka+1]   (ISA 16x4 f32 layout)
    a.x = (ka     < DESC1) ? desc[(rb + nl) * DESC1 + ka]     : 0.f;
    a.y = (ka + 1 < DESC1) ? desc[(rb + nl) * DESC1 + ka + 1] : 0.f;
    // B frag: lane holds B[ka][n=nn], B[ka+1][n=nn]
    bv.x = (ka     < DESC1) ? W1[ka * PH + nn]       : 0.f;
    bv.y = (ka + 1 < DESC1) ? W1[(ka + 1) * PH + nn] : 0.f;
    c = __builtin_amdgcn_wmma_f32_16x16x4_f32(false, a, false, bv, (short)0, c,
                                              false, false);
  }
  #pragma unroll
  for (int v = 0; v < 8; ++v) {
    int m = v + 8 * kh;                 // C/D layout: row = v + 8*(lane/16)
    hout[m * PH + nn] = c[v] + b1[nn];
  }
  __syncthreads();

  if (tid < 16) {                       // LN stats, one thread per row
    float s = 0.f;
    for (int j = 0; j < PH; ++j) s += hout[tid * PH + j];
    float muv = s / (float)PH;
    float vs = 0.f;
    for (int j = 0; j < PH; ++j) { float d = hout[tid * PH + j] - muv; vs += d * d; }
    mrow[tid] = muv;
    rstd[tid] = 1.f / sqrtf(vs / (float)PH + 1e-5f);
  }
  __syncthreads();
  for (int idx = tid; idx < 16 * PH; idx += TPB) {
    int m = idx >> 7, j = idx & (PH - 1);
    float xh = (hout[idx] - mrow[m]) * rstd[m] * g1[j] + bb1[j];
    float ge = 0.5f * xh * (1.f + erff(xh * 0.70710678118654752f)); // exact GELU
    h[(rb + m) * PH + j] = ge;
  }
}

// ---------------------------------------------------------------------------
// Kernel 3: GEMM2 (1024x128 @ 128x4096) via f32 WMMA; h-block staged in LDS.
// Grid (64, 32): 16 rows x 128 cols per block; 32 wmma steps per wave.
// Writes pre-LN result (+bias) into d_out.
// ---------------------------------------------------------------------------
__global__ __launch_bounds__(TPB)
void mlp2_kernel(const float* __restrict__ h, const float* __restrict__ W2,
                 const float* __restrict__ b2, float* __restrict__ out) {
  __shared__ float hs[16 * PH];
  const int tid = threadIdx.x;
  const int lane = tid & 31, wave = tid >> 5;
  const int nl = lane & 15, kh = lane >> 4;
  const int rb = blockIdx.x * 16;
  const int cb = blockIdx.y * 128;
  for (int idx = tid; idx < 16 * PH; idx += TPB)
    hs[idx] = h[(rb + (idx >> 7)) * PH + (idx & (PH - 1))];
  __syncthreads();

  const int nn = cb + wave * 16 + nl;
  v8f c = {0.f, 0.f, 0.f, 0.f, 0.f, 0.f, 0.f, 0.f};
  for (int kb = 0; kb < PH; kb += 4) {
    int ka = kb + 2 * kh;
    v2f a, bv;
    a.x  = hs[nl * PH + ka];
    a.y  = hs[nl * PH + ka + 1];
    bv.x = W2[(size_t)ka * HIDn + nn];
    bv.y = W2[(size_t)(ka + 1) * HIDn + nn];
    c = __builtin_amdgcn_wmma_f32_16x16x4_f32(false, a, false, bv, (short)0, c,
                                              false, false);
  }
  #pragma unroll
  for (int v = 0; v < 8; ++v) {
    int m = v + 8 * kh;
    out[(size_t)(rb + m) * HIDn + nn] = c[v] + b2[nn];
  }
}

// ---------------------------------------------------------------------------
// Kernel 4: in-place LayerNorm over 4096 per row (deterministic two-pass,
// row cached in registers before any write) + writes the all-ones mask.
// ---------------------------------------------------------------------------
__global__ __launch_bounds__(TPB)
void ln2_kernel(const float* __restrict__ g2, const float* __restrict__ bb2,
                float* __restrict__ out, float* __restrict__ maskout) {
  __shared__ float wred[NWAVE];
  const int tid = threadIdx.x;
  const int lane = tid & 31, wave = tid >> 5;
  const int row = blockIdx.x;
  float x[HIDn / TPB];                 // 16 cached elements per thread
  float s = 0.f;
  #pragma unroll
  for (int i = 0; i < HIDn / TPB; ++i) {
    x[i] = out[(size_t)row * HIDn + tid + i * TPB];
    s += x[i];
  }
  s += __shfl_down(s, 16, 32); s += __shfl_down(s, 8, 32);
  s += __shfl_down(s, 4, 32);  s += __shfl_down(s, 2, 32);
  s += __shfl_down(s, 1, 32);
  if (lane == 0) wred[wave] = s;
  __syncthreads();
  float tot = 0.f;
  for (int w = 0; w < NWAVE; ++w) tot += wred[w];
  const float muv = tot / (float)HIDn;
  __syncthreads();

  float vs = 0.f;
  #pragma unroll
  for (int i = 0; i < HIDn / TPB; ++i) { float d = x[i] - muv; vs += d * d; }
  vs += __shfl_down(vs, 16, 32); vs += __shfl_down(vs, 8, 32);
  vs += __shfl_down(vs, 4, 32);  vs += __shfl_down(vs, 2, 32);
  vs += __shfl_down(vs, 1, 32);
  if (lane == 0) wred[wave] = vs;
  __syncthreads();
  float vtot = 0.f;
  for (int w = 0; w < NWAVE; ++w) vtot += wred[w];
  const float rstd = 1.f / sqrtf(vtot / (float)HIDn + 1e-5f);

  #pragma unroll
  for (int i = 0; i < HIDn / TPB; ++i) {
    int j = tid + i * TPB;
    out[(size_t)row * HIDn + j] = (x[i] - muv) * rstd * g2[j] + bb2[j];
  }
  if (tid == 0) maskout[row] = 1.f;
}

// ---------------------------------------------------------------------------
extern "C" void kernel_launch(void* const* d_in, const int* in_sizes, int n_in,
                              void* d_out, int out_size, void* d_ws, size_t ws_size,
                              hipStream_t stream) {
  const float* fix = (const float*)d_in[0];
  const float* W1  = (const float*)d_in[1];
  const float* b1  = (const float*)d_in[2];
  const float* g1  = (const float*)d_in[3];
  const float* bb1 = (const float*)d_in[4];
  const float* W2  = (const float*)d_in[5];
  const float* b2  = (const float*)d_in[6];
  const float* g2  = (const float*)d_in[7];
  const float* bb2 = (const float*)d_in[8];
  const int* attn  = (const int*)d_in[9];

  float* out  = (float*)d_out;
  float* desc = (float*)d_ws;                 // B*34 floats
  float* h    = desc + (size_t)Bn * DESC1;    // B*128 floats

  const size_t emlds =
      (size_t)(Sn * Fn + Sn + Sn * Kn + NWAVE * 18) * sizeof(float); // ~148 KB
  (void)hipFuncSetAttribute((const void*)gmm_em_kernel,
                            hipFuncAttributeMaxDynamicSharedMemorySize,
                            (int)emlds);

  gmm_em_kernel<<<Bn, TPB, emlds, stream>>>(fix, attn, desc);
  mlp1_kernel<<<Bn / 16, TPB, 0, stream>>>(desc, W1, b1, g1, bb1, h);
  dim3 gg(Bn / 16, HIDn / 128);
  mlp2_kernel<<<gg, TPB, 0, stream>>>(h, W2, b2, out);
  ln2_kernel<<<Bn, TPB, 0, stream>>>(g2, bb2, out, out + (size_t)Bn * HIDn);
}